// RandomAttention_20830591386186
// MI455X (gfx1250) — compile-verified
//
#include <hip/hip_runtime.h>

// Random attention: B=2, S=2048, NH=8, H=64, NKEYS=64, fp32.
// One workgroup per (b, q); wave w handles head w. K/V gathered rows staged in
// LDS; both matvecs done with full-precision V_WMMA_F32_16X16X4_F32.

#define B_   2
#define S_   2048
#define NH_  8
#define H_   64
#define NK_  64
#define STRIDE 68   // padded LDS row stride (floats): 272B rows -> 16B-aligned float4 slots

typedef __attribute__((ext_vector_type(2))) float v2f;
typedef __attribute__((ext_vector_type(8))) float v8f;

__global__ __launch_bounds__(256, 1)
void rand_attn_wmma(const float* __restrict__ qg,
                    const float* __restrict__ kg,
                    const float* __restrict__ vg,
                    const int*   __restrict__ idxg,
                    float*       __restrict__ outg)
{
    extern __shared__ unsigned char smem_raw[];
    float* kvbuf = (float*)smem_raw;                 // [NH][64*STRIDE] gathered K, then V
    float* scbuf = kvbuf + NH_ * 64 * STRIDE;        // [NH][64] scores -> softmax weights
    float* qbuf  = scbuf + NH_ * 64;                 // [NH][64] q, later reused for z
    int*   ibuf  = (int*)(qbuf + NH_ * 64);          // [64] gather indices (uniform per block)

    const int tid  = threadIdx.x;
    const int lane = tid & 31;
    const int hd   = tid >> 5;        // head = wave id (8 waves)
    const int li   = lane & 15;
    const int hi   = lane >> 4;       // half-wave select (ISA K-split)

    const int bid = blockIdx.x;       // bid = b*S + sq
    const int b   = bid >> 11;        // S = 2048
    const int sq  = bid & (S_ - 1);

    // indices are shared by all 8 heads of this (b, q)
    if (tid < NK_) ibuf[tid] = idxg[(size_t)bid * NK_ + tid];
    __syncthreads();

    float* kv = kvbuf + hd * (64 * STRIDE);
    float* sc = scbuf + hd * 64;
    float* qe = qbuf  + hd * 64;

    const size_t rowbase = ((size_t)(b * S_ + sq) * NH_ + hd) * H_;
    *(v2f*)&qe[2 * lane] = *(const v2f*)&qg[rowbase + 2 * lane];

    // ---- gather K rows (head slice is a contiguous 256B chunk) ----
    for (int j = 0; j < 32; ++j) {
        int row  = 2 * j + hi;
        int krow = ibuf[row];
        const float4* src = (const float4*)&kg[((size_t)(b * S_ + krow) * NH_ + hd) * H_];
        float4 d = src[li];
        *(float4*)&kv[row * STRIDE + 4 * li] = d;
    }

    // ---- scores = Kg(64x64) * q(64) via chained f32 WMMA, K-chunks of 4 ----
    // A 16x4 layout: lanes0-15: M=lane, v0/v1 = K0/K1; lanes16-31: M=lane-16, v0/v1 = K2/K3.
    // B 4x16 layout: lanes hold N; lanes0-15 v0/v1 = K0/K1 rows, lanes16-31 = K2/K3.
    // We broadcast q across all 16 N columns -> every D column equals the score vector.
    v8f a0 = {0,0,0,0,0,0,0,0}, a1 = a0, a2 = a0, a3 = a0;
    for (int t = 0; t < 16; ++t) {
        v2f bv = *(const v2f*)&qe[4 * t + 2 * hi];
        v2f m0 = *(const v2f*)&kv[( 0 + li) * STRIDE + 4 * t + 2 * hi];
        v2f m1 = *(const v2f*)&kv[(16 + li) * STRIDE + 4 * t + 2 * hi];
        v2f m2 = *(const v2f*)&kv[(32 + li) * STRIDE + 4 * t + 2 * hi];
        v2f m3 = *(const v2f*)&kv[(48 + li) * STRIDE + 4 * t + 2 * hi];
        a0 = __builtin_amdgcn_wmma_f32_16x16x4_f32(false, m0, false, bv, (short)0, a0, false, false);
        a1 = __builtin_amdgcn_wmma_f32_16x16x4_f32(false, m1, false, bv, (short)0, a1, false, false);
        a2 = __builtin_amdgcn_wmma_f32_16x16x4_f32(false, m2, false, bv, (short)0, a2, false, false);
        a3 = __builtin_amdgcn_wmma_f32_16x16x4_f32(false, m3, false, bv, (short)0, a3, false, false);
    }
    // C/D layout: VGPR r -> row r (lanes0-15) / row r+8 (lanes16-31); col = lane%16.
    if (li == 0) {
#pragma unroll
        for (int r = 0; r < 8; ++r) {
            sc[ 0 + 8 * hi + r] = a0[r];
            sc[16 + 8 * hi + r] = a1[r];
            sc[32 + 8 * hi + r] = a2[r];
            sc[48 + 8 * hi + r] = a3[r];
        }
    }

    // ---- softmax over the 64 keys (wave-local; lane owns keys lane, lane+32) ----
    float s0 = sc[lane]      * 0.125f;   // scale = H^-0.5 = 1/8
    float s1 = sc[lane + 32] * 0.125f;
    float mx = fmaxf(s0, s1);
#pragma unroll
    for (int off = 16; off >= 1; off >>= 1)
        mx = fmaxf(mx, __shfl_xor(mx, off, 32));
    float e0 = __expf(s0 - mx), e1 = __expf(s1 - mx);
    float sm = e0 + e1;
#pragma unroll
    for (int off = 16; off >= 1; off >>= 1)
        sm += __shfl_xor(sm, off, 32);
    float inv = 1.0f / sm;
    sc[lane]      = e0 * inv;
    sc[lane + 32] = e1 * inv;

    // ---- gather V rows into the same per-head LDS buffer (wave DS ops are in order) ----
    for (int j = 0; j < 32; ++j) {
        int row  = 2 * j + hi;
        int krow = ibuf[row];
        const float4* src = (const float4*)&vg[((size_t)(b * S_ + krow) * NH_ + hd) * H_];
        float4 d = src[li];
        *(float4*)&kv[row * STRIDE + 4 * li] = d;
    }

    // ---- z^T = Vg^T(64h x 64k) * w(64) via the same WMMA matvec trick ----
    a0 = a1 = a2 = a3 = (v8f){0,0,0,0,0,0,0,0};
    for (int t = 0; t < 16; ++t) {
        int kk = 4 * t + 2 * hi;                  // this half-wave's two K (=key) indices
        v2f bv = *(const v2f*)&sc[kk];
        v2f m0, m1, m2, m3;
        m0.x = kv[kk * STRIDE +  0 + li];  m0.y = kv[(kk + 1) * STRIDE +  0 + li];
        m1.x = kv[kk * STRIDE + 16 + li];  m1.y = kv[(kk + 1) * STRIDE + 16 + li];
        m2.x = kv[kk * STRIDE + 32 + li];  m2.y = kv[(kk + 1) * STRIDE + 32 + li];
        m3.x = kv[kk * STRIDE + 48 + li];  m3.y = kv[(kk + 1) * STRIDE + 48 + li];
        a0 = __builtin_amdgcn_wmma_f32_16x16x4_f32(false, m0, false, bv, (short)0, a0, false, false);
        a1 = __builtin_amdgcn_wmma_f32_16x16x4_f32(false, m1, false, bv, (short)0, a1, false, false);
        a2 = __builtin_amdgcn_wmma_f32_16x16x4_f32(false, m2, false, bv, (short)0, a2, false, false);
        a3 = __builtin_amdgcn_wmma_f32_16x16x4_f32(false, m3, false, bv, (short)0, a3, false, false);
    }
    if (li == 0) {                       // stash z through LDS for coalesced store
#pragma unroll
        for (int r = 0; r < 8; ++r) {
            qe[ 0 + 8 * hi + r] = a0[r];
            qe[16 + 8 * hi + r] = a1[r];
            qe[32 + 8 * hi + r] = a2[r];
            qe[48 + 8 * hi + r] = a3[r];
        }
    }
    v2f z = *(const v2f*)&qe[2 * lane];
    *(v2f*)&outg[rowbase + 2 * lane] = z;
}

extern "C" void kernel_launch(void* const* d_in, const int* in_sizes, int n_in,
                              void* d_out, int out_size, void* d_ws, size_t ws_size,
                              hipStream_t stream) {
    (void)in_sizes; (void)n_in; (void)out_size; (void)d_ws; (void)ws_size;
    const float* q   = (const float*)d_in[0];
    const float* k   = (const float*)d_in[1];
    const float* v   = (const float*)d_in[2];
    const int*   idx = (const int*)d_in[3];
    float* out = (float*)d_out;

    const size_t smem = (size_t)(NH_ * 64 * STRIDE + NH_ * 64 + NH_ * 64) * sizeof(float)
                      + (size_t)NK_ * sizeof(int);   // 143,616 B (two blocks per 320KB WGP)
    dim3 grid(B_ * S_);
    dim3 block(256);
    hipLaunchKernelGGL(rand_attn_wmma, grid, block, smem, stream, q, k, v, idx, out);
}